// HarmonicCQT_82016695484690
// MI455X (gfx1250) — compile-verified
//
#include <hip/hip_runtime.h>
#include <stdint.h>

typedef float v2f __attribute__((ext_vector_type(2)));
typedef float v8f __attribute__((ext_vector_type(8)));

#define HOP      512
#define NBINS    84
#define NCOL     168   // real + imag columns stacked
#define WSTRIDE  192   // padded N (12 tiles * 16): every wave owns 3 tiles
#define BATCH    4
#define LPAD     3     // LDS pad floats per 512-float chunk (bank de-conflict)

// ---------------------------------------------------------------------------
// One-shot weight transpose into k-pair-interleaved layout:
//   Wt[(k>>1)*2*WSTRIDE + 2n + (k&1)] = (n<84 ? kr[n][k] : n<168 ? ki[n-84][k] : 0)
// so a lane's (r0, r0+1) B pair is one contiguous 8B -> single global_load_b64,
// coalesced across lanes (consecutive n -> 8B stride).
// ---------------------------------------------------------------------------
__global__ void cqt_transpose_weights(const float* __restrict__ kr,
                                      const float* __restrict__ ki,
                                      float* __restrict__ wt,
                                      int Klen, int Kpad) {
    int idx = blockIdx.x * blockDim.x + threadIdx.x;
    int total = Kpad * WSTRIDE;
    if (idx >= total) return;
    int k = idx / WSTRIDE;
    int n = idx % WSTRIDE;
    float v = 0.0f;
    if (k < Klen) {
        if (n < NBINS)     v = kr[(size_t)n * Klen + k];
        else if (n < NCOL) v = ki[(size_t)(n - NBINS) * Klen + k];
    }
    wt[(size_t)(k >> 1) * (2 * WSTRIDE) + 2 * n + (k & 1)] = v;
}

__device__ inline float fetchW(const float* __restrict__ kr,
                               const float* __restrict__ ki,
                               int n, int k, int Klen) {
    if (k >= Klen) return 0.0f;
    if (n < NBINS) return kr[(size_t)n * Klen + k];
    if (n < NCOL)  return ki[(size_t)(n - NBINS) * Klen + k];
    return 0.0f;
}

__device__ inline v8f wmma4(v2f a, v2f b, v8f c) {
    // D = A(16x4 f32) * B(4x16 f32) + C(16x16 f32)
    return __builtin_amdgcn_wmma_f32_16x16x4_f32(
        /*neg_a=*/false, a, /*neg_b=*/false, b,
        /*c_mod=*/(short)0, c, /*reuse_a=*/false, /*reuse_b=*/false);
}

__device__ inline void storeTile(float* __restrict__ out, v8f c, int nt,
                                 int nl, int rowhalf8, int mt, int b,
                                 int nblocks) {
    const int n = nt * 16 + nl;
    if (n >= NCOL) return;                       // pad columns (tiles 10/11)
    const int bin  = (n < NBINS) ? n : n - NBINS;
    const int reim = (n < NBINS) ? 0 : 1;
    float* op = out + ((size_t)b * nblocks * NBINS + bin) * 2 + reim;
#pragma unroll
    for (int r = 0; r < 8; ++r) {
        const int t = mt * 16 + rowhalf8 + r;    // D vgpr r: row = r + 8*(lane>>4)
        if (t < nblocks)
            op[(size_t)t * (NBINS * 2)] = c[r];
    }
}

// ---------------------------------------------------------------------------
// Main GEMM: grid = (M-tiles, batch). 128 threads = 4 wave32.
// Wave w accumulates N-tiles {w, w+4, w+8} unconditionally (tile 11 is pad).
// LDS: padded-linear audio span, sample s -> lds[s + (s>>9)*LPAD]; A lanes
// read at stride 515 floats -> conflict-free banks, linear hot-loop address.
// Inner loop batches 4 K-steps: 16 loads issued, then 12 WMMAs consume them,
// letting counter-based waits overlap loads with matrix ops.
// ---------------------------------------------------------------------------
template<bool TRANS>
__global__ __launch_bounds__(128)
void cqt_wmma(const float* __restrict__ audio,
              const float* __restrict__ wt,
              const float* __restrict__ kr,
              const float* __restrict__ ki,
              float* __restrict__ out,
              int T, int Klen, int Kpad, int nblocks) {
    extern __shared__ float lds[];

    const int mt = blockIdx.x;
    const int b  = blockIdx.y;
    const int span = 15 * HOP + Kpad;
    const long sampStart = (long)mt * 16 * HOP;
    const float* aud = audio + (size_t)b * T;

    // ---- Stage audio span into LDS via CDNA5 async-to-LDS (ASYNCcnt). ----
    for (int i = threadIdx.x; i < span; i += blockDim.x) {
        const long pos = sampStart + i;
        const int phys = i + (i >> 9) * LPAD;
        if (pos < (long)T) {
            const float* gp = aud + pos;
            unsigned ldsoff = (unsigned)(uintptr_t)(lds + phys);
            asm volatile("global_load_async_to_lds_b32 %0, %1, off"
                         :: "v"(ldsoff), "v"(gp)
                         : "memory");
        } else {
            lds[phys] = 0.0f;   // zero-padded tail lanes, normal DS path
        }
    }
    asm volatile("s_wait_asynccnt 0x0" ::: "memory");
    __syncthreads();

    const int lane = threadIdx.x & 31;
    const int wave = threadIdx.x >> 5;
    const int m    = lane & 15;          // A-fragment row
    const int h1   = lane >> 4;          // half-select
    const int h2   = h1 << 1;            // K sub-offset: 0 or 2
    const int nl   = lane & 15;          // B/D column within tile
    const int n0   = wave * 16 + nl;

    // phys(m*512 + kk + h2) = m*(512+LPAD) + h2  +  kk + (kk>>9)*LPAD
    const int aoff = m * (HOP + LPAD) + h2;            // lane part
    const int boff = h1 * (2 * WSTRIDE) + n0 * 2;      // lane part (floats)

    v8f acc0 = {}; v8f acc1 = {}; v8f acc2 = {};

    int kbase = 0;                                     // uniform phys base
    for (int kq = 0; kq < Kpad; kq += HOP) {
        const int kcnt = (Kpad - kq < HOP) ? (Kpad - kq) : HOP;   // mult of 16
        const float* abase = lds + aoff + kbase;                  // DS imm offs
        const float* wbase = wt + (size_t)(kq >> 1) * (2 * WSTRIDE) + boff;
        for (int k2 = 0; k2 < kcnt; k2 += 16) {
            v2f a[4], b0[4], b1[4], b2[4];
#pragma unroll
            for (int u = 0; u < 4; ++u) {
                const int kk2 = k2 + 4 * u;
                a[u].x = abase[kk2];
                a[u].y = abase[kk2 + 1];
                if (TRANS) {
                    const float* pk = wbase + (size_t)(kk2 >> 1) * (2 * WSTRIDE);
                    b0[u] = *(const v2f*)(pk);         // tile w      (b64)
                    b1[u] = *(const v2f*)(pk + 128);   // tile w+4
                    b2[u] = *(const v2f*)(pk + 256);   // tile w+8
                } else {
                    const int r0 = kq + kk2 + h2;
                    b0[u].x = fetchW(kr, ki, n0,       r0,     Klen);
                    b0[u].y = fetchW(kr, ki, n0,       r0 + 1, Klen);
                    b1[u].x = fetchW(kr, ki, n0 + 64,  r0,     Klen);
                    b1[u].y = fetchW(kr, ki, n0 + 64,  r0 + 1, Klen);
                    b2[u].x = fetchW(kr, ki, n0 + 128, r0,     Klen);
                    b2[u].y = fetchW(kr, ki, n0 + 128, r0 + 1, Klen);
                }
            }
#pragma unroll
            for (int u = 0; u < 4; ++u) {
                acc0 = wmma4(a[u], b0[u], acc0);
                acc1 = wmma4(a[u], b1[u], acc1);
                acc2 = wmma4(a[u], b2[u], acc2);
            }
        }
        kbase += HOP + LPAD;
    }

    // ---- Store D tiles (pad tiles masked inside) ----
    const int rowhalf8 = h1 * 8;
    storeTile(out, acc0, wave,     nl, rowhalf8, mt, b, nblocks);
    storeTile(out, acc1, wave + 4, nl, rowhalf8, mt, b, nblocks);
    storeTile(out, acc2, wave + 8, nl, rowhalf8, mt, b, nblocks);
}

// ---------------------------------------------------------------------------
extern "C" void kernel_launch(void* const* d_in, const int* in_sizes, int n_in,
                              void* d_out, int out_size, void* d_ws, size_t ws_size,
                              hipStream_t stream) {
    (void)n_in; (void)out_size;
    const float* audio = (const float*)d_in[0];
    const float* kr    = (const float*)d_in[1];
    const float* ki    = (const float*)d_in[2];
    float* out = (float*)d_out;

    const int T       = in_sizes[0] / BATCH;       // C == 1
    const int Klen    = in_sizes[1] / NBINS;       // 11341
    const int Kpad    = (Klen + 15) & ~15;         // multiple of 16 (batched steps)
    const int nblocks = T / HOP + 1;               // 1292
    const int mtiles  = (nblocks + 15) / 16;       // 81
    const int span    = 15 * HOP + Kpad;           // audio floats per M-tile
    const int chunks  = (span + HOP - 1) / HOP;
    const size_t ldsBytes = (size_t)(span + chunks * LPAD) * sizeof(float); // ~75 KB
    const size_t wtBytes  = (size_t)Kpad * WSTRIDE * sizeof(float);         // ~8.3 MB

    dim3 grid(mtiles, BATCH);
    float* wtp = (float*)d_ws;

    if (ws_size >= wtBytes) {
        const int total = Kpad * WSTRIDE;
        cqt_transpose_weights<<<(total + 255) / 256, 256, 0, stream>>>(
            kr, ki, wtp, Klen, Kpad);
        cqt_wmma<true><<<grid, 128, ldsBytes, stream>>>(
            audio, wtp, kr, ki, out, T, Klen, Kpad, nblocks);
    } else {
        cqt_wmma<false><<<grid, 128, ldsBytes, stream>>>(
            audio, wtp, kr, ki, out, T, Klen, Kpad, nblocks);
    }
}